// SwitchFocusedLossAdditiveNoSegmentation_80221399155358
// MI455X (gfx1250) — compile-verified
//
#include <hip/hip_runtime.h>

// ---------------------------------------------------------------------------
// SwitchFocusedLoss for MI455X (gfx1250, wave32).
// Memory-bound: ~42 MB read once -> ~1.8 us at 23.3 TB/s HBM.
// Round 2 changes: split each row into 8 segments (1024 blocks) to fill the
// whole chip's memory concurrency; exact +/-5 window across segment edges via
// 5-position halo bits; fully unrolled 8-iteration load loop for deep MLP.
// Final 128-row combine + 1024-segment reduction via v_wmma_f32_16x16x4_f32.
// ---------------------------------------------------------------------------

typedef __attribute__((ext_vector_type(2))) float v2f;
typedef __attribute__((ext_vector_type(8))) float v8f;

#define SEQ     16384
#define NROWS   128
#define NSEG    8
#define SEGLEN  (SEQ / NSEG)        // 2048 positions per segment
#define LWORDS  (SEGLEN / 32)       // 64 bitmask words per segment
#define NSEGTOT (NROWS * NSEG)      // 1024 segments total
#define BLK     256
#define ITERS   (SEGLEN / BLK)      // 8
#define TOL     5

// windowed-any over +/-5 bit positions; zero words give clamped edges
__device__ __forceinline__ unsigned window_or(unsigned prev, unsigned cur, unsigned next) {
  unsigned long long lo = ((unsigned long long)cur  << 32) | (unsigned long long)prev;
  unsigned long long hi = ((unsigned long long)next << 32) | (unsigned long long)cur;
  unsigned r = cur;
#pragma unroll
  for (int d = 1; d <= TOL; ++d) {
    r |= (unsigned)(hi >> d);         // bit p gets mask(p+d)
    r |= (unsigned)(lo >> (32 - d));  // bit p gets mask(p-d)
  }
  return r;
}

__device__ __forceinline__ int argmax4(const float4 lg) {
  float m = lg.x; int pred = 0;
  if (lg.y > m) { m = lg.y; pred = 1; }
  if (lg.z > m) { m = lg.z; pred = 2; }
  if (lg.w > m) { m = lg.w; pred = 3; }
  return pred;
}

__global__ __launch_bounds__(BLK)
void seg_loss_kernel(const float4* __restrict__ logits,
                     const int*    __restrict__ labels,
                     float*        __restrict__ ws) {
  const int blk  = blockIdx.x;
  const int row  = blk / NSEG;
  const int seg  = blk % NSEG;
  const int tid  = threadIdx.x;
  const int lane = tid & 31;
  const int wave = tid >> 5;
  const int seg0 = seg * SEGLEN;

  // bitmask arrays with one halo word on each side (indices 0 and LWORDS+1)
  __shared__ unsigned predBits[LWORDS + 2];
  __shared__ unsigned trueBits[LWORDS + 2];
  __shared__ unsigned validBits[LWORDS];
  __shared__ unsigned hasTrueFlag;
  __shared__ float               redF[BLK];
  __shared__ unsigned long long  redL[BLK];

  if (tid < 2) {                       // zero the two halo words
    const int hw = tid * (LWORDS + 1);
    predBits[hw] = 0u;
    trueBits[hw] = 0u;
  }
  if (tid == 0) hasTrueFlag = 0u;
  __syncthreads();

  const size_t rowOff = (size_t)row * SEQ;
  const float4* __restrict__ lrow   = logits + rowOff;
  const int*    __restrict__ labrow = labels + rowOff;

  float    baseSum = 0.0f;
  unsigned trueAny = 0u;

#pragma unroll
  for (int it = 0; it < ITERS; ++it) {
    const int pos = seg0 + it * BLK + tid;   // coalesced b128 + b32 loads
    const float4 lg  = lrow[pos];
    const int    lbl = labrow[pos];

    const bool valid = (lbl != -100);
    const int  y     = valid ? lbl : 0;
    const int  pred  = argmax4(lg);

    float m = fmaxf(fmaxf(lg.x, lg.y), fmaxf(lg.z, lg.w));
    const float se   = __expf(lg.x - m) + __expf(lg.y - m)
                     + __expf(lg.z - m) + __expf(lg.w - m);
    const float logZ = m + __logf(se);
    const float xy   = (y == 0) ? lg.x : (y == 1) ? lg.y : (y == 2) ? lg.z : lg.w;
    const float wy   = (y < 2) ? 0.1f : 5.0f;     // CLASS_WEIGHTS
    baseSum += valid ? wy * (logZ - xy) : 0.0f;

    // pack 32 positions/word via wave32 ballot (lane order == position order)
    const unsigned pm = (unsigned)__ballot(pred >= 2);
    const unsigned tm = (unsigned)__ballot(lbl  >= 2);
    const unsigned vm = (unsigned)__ballot(valid);
    if (lane == 0) {
      const int w = it * 8 + wave;             // local word index
      predBits[w + 1]  = pm;
      trueBits[w + 1]  = tm;
      validBits[w]     = vm;
    }
    trueAny |= tm;
  }

  // halo bits: the TOL positions just outside each segment edge (clamped to row)
  if (tid < 2 * TOL) {
    const bool left  = tid < TOL;
    const int  p     = left ? (seg0 - TOL + tid) : (seg0 + SEGLEN + (tid - TOL));
    const bool inRow = left ? (seg0 > 0) : (seg0 + SEGLEN < SEQ);
    if (inRow) {
      const float4 lg  = lrow[p];
      const int    lbl = labrow[p];
      const int    pred = argmax4(lg);
      const unsigned bit = 1u << (p & 31);
      const int      w   = left ? 0 : (LWORDS + 1);
      if (pred >= 2) atomicOr(&predBits[w], bit);
      if (lbl  >= 2) atomicOr(&trueBits[w], bit);
    }
  }
  if (lane == 0 && trueAny) atomicOr(&hasTrueFlag, 1u);
  __syncthreads();

  // bitwise windowed-any + popcount counting
  int rewardCnt = 0, penaltyCnt = 0, validCnt = 0;
  for (int w = tid; w < LWORDS; w += BLK) {
    const unsigned predNear = window_or(predBits[w], predBits[w + 1], predBits[w + 2]);
    const unsigned trueNear = window_or(trueBits[w], trueBits[w + 1], trueBits[w + 2]);
    const unsigned vm       = validBits[w];
    rewardCnt  += __popc(vm & trueBits[w + 1] & predNear);
    penaltyCnt += __popc(vm & predBits[w + 1] & ~trueNear);
    validCnt   += __popc(vm);
  }

  // deterministic block reduction: 1 float + 3 counts packed in one u64
  redF[tid] = baseSum;
  redL[tid] = (unsigned long long)rewardCnt
            | ((unsigned long long)penaltyCnt << 21)
            | ((unsigned long long)validCnt   << 42);
  __syncthreads();
  for (int s = BLK / 2; s > 0; s >>= 1) {
    if (tid < s) {
      redF[tid] += redF[tid + s];
      redL[tid] += redL[tid + s];
    }
    __syncthreads();
  }

  if (tid == 0) {
    const unsigned long long p = redL[0];
    ws[blk]               = redF[0];                                  // base-loss sum
    ws[NSEGTOT     + blk] = (float)(int)( p        & 0x1FFFFFull);    // reward count
    ws[2 * NSEGTOT + blk] = (float)(int)((p >> 21) & 0x1FFFFFull);    // penalty count (ungated)
    ws[3 * NSEGTOT + blk] = (float)(int)((p >> 42) & 0x1FFFFFull);    // valid count
    ws[4 * NSEGTOT + blk] = hasTrueFlag ? 1.0f : 0.0f;                // segment true-any
  }
}

// Single-wave final reduction. Each lane folds 4 rows (8 segments each,
// applying the per-row has_true gate), producing the two v2f B-chunks for the
// layout-robust WMMA full-sum: with A = all-ones, sum_{n<16} D[0][n] equals
// the sum of ALL 64 B entries regardless of B's VGPR mapping.
__global__ __launch_bounds__(32)
void final_reduce_wmma(const float* __restrict__ ws, float* __restrict__ out) {
  const int lane = threadIdx.x;   // 0..31, EXEC all ones for WMMA

  float adj[4], cnt[4];
#pragma unroll
  for (int q = 0; q < 4; ++q) {
    const int r = lane + q * 32;                 // row index
    float base = 0.f, rew = 0.f, pen = 0.f, c = 0.f, any = 0.f;
#pragma unroll
    for (int s = 0; s < NSEG; ++s) {
      const int g = r * NSEG + s;
      base += ws[g];
      rew  += ws[NSEGTOT     + g];
      pen  += ws[2 * NSEGTOT + g];
      c    += ws[3 * NSEGTOT + g];
      any  += ws[4 * NSEGTOT + g];
    }
    adj[q] = base - 2.0f * rew + ((any > 0.0f) ? 1.5f * pen : 0.0f);
    cnt[q] = c;
  }

  v2f ones; ones[0] = 1.0f; ones[1] = 1.0f;
  v2f a0, a1, c0, c1;
  a0[0] = adj[0]; a0[1] = adj[1];   // rows 0..63 across the wave
  a1[0] = adj[2]; a1[1] = adj[3];   // rows 64..127
  c0[0] = cnt[0]; c0[1] = cnt[1];
  c1[0] = cnt[2]; c1[1] = cnt[3];

  v8f accA = {};
  accA = __builtin_amdgcn_wmma_f32_16x16x4_f32(false, ones, false, a0, (short)0, accA, false, false);
  accA = __builtin_amdgcn_wmma_f32_16x16x4_f32(false, ones, false, a1, (short)0, accA, false, false);
  v8f accC = {};
  accC = __builtin_amdgcn_wmma_f32_16x16x4_f32(false, ones, false, c0, (short)0, accC, false, false);
  accC = __builtin_amdgcn_wmma_f32_16x16x4_f32(false, ones, false, c1, (short)0, accC, false, false);

  __shared__ float sA[32], sC[32];
  sA[lane] = accA[0];   // lanes 0..15 hold D[0][n] = column sums
  sC[lane] = accC[0];
  __syncthreads();

  if (lane == 0) {
    float ta = 0.0f, tc = 0.0f;
    for (int i = 0; i < 16; ++i) { ta += sA[i]; tc += sC[i]; }
    out[0] = ta / tc;   // masked mean
  }
}

extern "C" void kernel_launch(void* const* d_in, const int* in_sizes, int n_in,
                              void* d_out, int out_size, void* d_ws, size_t ws_size,
                              hipStream_t stream) {
  const float4* logits = (const float4*)d_in[0];  // [128,16384,4] f32 -> one float4/position
  const int*    labels = (const int*)d_in[1];     // [128,16384] int
  float* ws  = (float*)d_ws;                      // 5*1024 floats = 20 KB scratch
  float* out = (float*)d_out;                     // scalar f32

  seg_loss_kernel<<<NSEGTOT, BLK, 0, stream>>>(logits, labels, ws);
  final_reduce_wmma<<<1, 32, 0, stream>>>(ws, out);
}